// NoSegAugmentedMemoryMultiheadAttentionBmm_12249246728830
// MI455X (gfx1250) — compile-verified
//
#include <hip/hip_runtime.h>
#include <hip/hip_bf16.h>

typedef __attribute__((ext_vector_type(2))) float v2f;
typedef __attribute__((ext_vector_type(8))) float v8f;

#define T_   799
#define B_   8
#define D_   512
#define H_   8
#define HD_  64
#define KL_  798
#define NREL 129
#define TB_  (T_*B_)          // 6392
#define KLB_ (KL_*B_)         // 6384

__device__ __forceinline__ v8f wmma_f32(v2f a, v2f b, v8f c) {
  // D = A(16x4,f32) * B(4x16,f32) + C(16x16,f32)
  return __builtin_amdgcn_wmma_f32_16x16x4_f32(false, a, false, b, (short)0, c, false, false);
}

// ---------------- Kernel 1a: Q projection, scale folded ----------------
// Each wave: 16 rows x 64 cols (4 N-tiles), A fragment reused 4x.
// q[t,b,e] = sum_d x[t,b,d]*wq[e,d] + bq[e];  qs[n=b*8+h][t][hd] = q*0.125
__global__ __launch_bounds__(256) void qproj_kernel(
    const float* __restrict__ x, const float* __restrict__ wq,
    const float* __restrict__ bq, float* __restrict__ qs) {
  int wave = threadIdx.x >> 5, lane = threadIdx.x & 31;
  int half = lane >> 4, l16 = lane & 15;
  int row0 = blockIdx.x * 16;
  int c0   = wave * 64 + l16;                      // 8 waves cover all 512 cols
  int r    = row0 + l16;
  const float* arow = x + (size_t)((r < TB_) ? r : (TB_ - 1)) * D_;
  const float* brow = wq + (size_t)c0 * D_;        // tiles at +0, +16*512, +32*512, +48*512
  v8f acc[4] = {{}, {}, {}, {}};
#pragma unroll 4
  for (int ks = 0; ks < 128; ++ks) {
    int kk = ks * 4 + half * 2;
    v2f a = *(const v2f*)(arow + kk);
    acc[0] = wmma_f32(a, *(const v2f*)(brow + kk), acc[0]);
    acc[1] = wmma_f32(a, *(const v2f*)(brow + 16 * D_ + kk), acc[1]);
    acc[2] = wmma_f32(a, *(const v2f*)(brow + 32 * D_ + kk), acc[2]);
    acc[3] = wmma_f32(a, *(const v2f*)(brow + 48 * D_ + kk), acc[3]);
  }
#pragma unroll
  for (int j = 0; j < 4; ++j) {
    int cc = c0 + j * 16;
    float bias = bq[cc];
    int h = cc >> 6, hd = cc & 63;
    for (int i = 0; i < 8; ++i) {
      int rr = row0 + i + half * 8;
      if (rr < TB_) {
        int t = rr >> 3, b = rr & 7;
        qs[(((size_t)(b * H_ + h)) * 800 + t) * 64 + hd] = (acc[j][i] + bias) * 0.125f;
      }
    }
  }
}

// ---------------- Kernel 1b: KV projection ----------------
// mem_rc row k: k<6 -> mems[k], else x[k-6].  K stored [n][k][hd]; V stored transposed [n][hd][k].
__global__ __launch_bounds__(256) void kvproj_kernel(
    const float* __restrict__ x, const float* __restrict__ mems,
    const float* __restrict__ wkv, const float* __restrict__ bkv,
    float* __restrict__ kws, float* __restrict__ vtw) {
  int wave = threadIdx.x >> 5, lane = threadIdx.x & 31;
  int half = lane >> 4, l16 = lane & 15;
  int row0 = blockIdx.x * 16;
  int c0   = blockIdx.y * 512 + wave * 64 + l16;   // cols 0..1023
  int r    = row0 + l16;                           // < 6384 always
  int kidx = r >> 3, b = r & 7;
  const float* arow = (kidx < 6) ? (mems + ((size_t)kidx * B_ + b) * D_)
                                 : (x + ((size_t)(kidx - 6) * B_ + b) * D_);
  const float* brow = wkv + (size_t)c0 * D_;
  v8f acc[4] = {{}, {}, {}, {}};
#pragma unroll 4
  for (int ks = 0; ks < 128; ++ks) {
    int kk = ks * 4 + half * 2;
    v2f a = *(const v2f*)(arow + kk);
    acc[0] = wmma_f32(a, *(const v2f*)(brow + kk), acc[0]);
    acc[1] = wmma_f32(a, *(const v2f*)(brow + 16 * D_ + kk), acc[1]);
    acc[2] = wmma_f32(a, *(const v2f*)(brow + 32 * D_ + kk), acc[2]);
    acc[3] = wmma_f32(a, *(const v2f*)(brow + 48 * D_ + kk), acc[3]);
  }
#pragma unroll
  for (int j = 0; j < 4; ++j) {
    int cc = c0 + j * 16;
    float bias = bkv[cc];
    for (int i = 0; i < 8; ++i) {
      int rr = row0 + i + half * 8;
      int k2 = rr >> 3, b2 = rr & 7;
      float val = acc[j][i] + bias;
      if (cc < 512) {
        int h = cc >> 6, hd = cc & 63;
        kws[(((size_t)(b2 * H_ + h)) * 800 + k2) * 64 + hd] = val;
      } else {
        int e2 = cc - 512, h = e2 >> 6, hd = e2 & 63;
        vtw[(((size_t)(b2 * H_ + h)) * 64 + hd) * 800 + k2] = val;
      }
    }
  }
}

// ---------------- Kernel 2: qr[n,t,r] = qh . rpe_k_emb[r] ----------------
__global__ __launch_bounds__(256) void qrpe_kernel(
    const float* __restrict__ qs, const float* __restrict__ rpek,
    float* __restrict__ qr) {
  int wave = threadIdx.x >> 5, lane = threadIdx.x & 31;
  int half = lane >> 4, l16 = lane & 15;
  int rowTile = blockIdx.x * 8 + wave;        // 3200 row tiles over 64*800 rows
  int row0 = rowTile * 16;
  int row  = row0 + l16;
  int col  = blockIdx.y * 16 + l16;           // 0..143
  // qs row t=799 is in-bounds pad (finite junk, never consumed); clamp col for rpek
  const float* aptr = qs + (size_t)row * 64;
  const float* bptr = rpek + (size_t)((col < NREL) ? col : (NREL - 1)) * 64;
  v8f acc = {};
#pragma unroll
  for (int ks = 0; ks < 16; ++ks) {
    int kk = ks * 4 + half * 2;
    v2f a = *(const v2f*)(aptr + kk);
    v2f b = *(const v2f*)(bptr + kk);
    acc = wmma_f32(a, b, acc);
  }
  if (col < NREL)
    for (int i = 0; i < 8; ++i)
      qr[(size_t)(row0 + i + half * 8) * 132 + col] = acc[i];
}

// ---------------- Kernel 3: fused attention (per n, 16-row tile) ----------------
__global__ __launch_bounds__(256) void attn_kernel(
    const float* __restrict__ qs, const float* __restrict__ kws,
    const float* __restrict__ vtw, const float* __restrict__ qr,
    const int* __restrict__ rpe, const unsigned char* __restrict__ amask,
    const int* __restrict__ lengths, const int* __restrict__ uptr,
    const float* __restrict__ rpev, float* __restrict__ attn) {
  __shared__ float lds[16][800];       // logits -> probs
  __shared__ float buckets[16][132];   // rpe-v buckets
  __shared__ float outb[16][64];       // PV accumulation
  __shared__ float red[16][16];
  __shared__ float rowmax[16], rowsum[16], rownz[16], rowthr[16], rowsum2[16];

  const float NEGINF = -__builtin_inff();
  int n = blockIdx.y, t0 = blockIdx.x * 16;
  int b = n >> 3, h = n & 7;
  int tid = threadIdx.x, wave = tid >> 5, lane = tid & 31;
  int half = lane >> 4, l16 = lane & 15;
  int rcb  = T_ - uptr[0] - 7;
  int klen = lengths[b] + 6 + rcb;

  // -------- phase 1: logits = (q*scale) @ k^T + qr[rpe], masked --------
  v2f afr[16];
  {
    // t=799 row is in-bounds pad (finite junk), its outputs are never stored
    const float* qrow = qs + ((size_t)n * 800 + (t0 + l16)) * 64;
#pragma unroll
    for (int ks = 0; ks < 16; ++ks)
      afr[ks] = *(const v2f*)(qrow + ks * 4 + half * 2);
  }
  for (int kb = wave; kb < 50; kb += 8) {
    v8f c = {};
    int kcol = kb * 16 + l16;
    // rows 798,799 are in-bounds pad; their logit columns get -inf below
    const float* krow = kws + ((size_t)n * 800 + kcol) * 64;
    if (kb + 8 < 50) __builtin_prefetch(krow + 128 * 64, 0, 0);  // next K-block (gfx1250 global_prefetch)
#pragma unroll
    for (int ks = 0; ks < 16; ++ks) {
      v2f bf = *(const v2f*)(krow + ks * 4 + half * 2);
      c = wmma_f32(afr[ks], bf, c);
    }
    int kc = (kcol < KL_) ? kcol : (KL_ - 1);
    for (int i = 0; i < 8; ++i) {
      int m = i + half * 8;
      int t = t0 + m;
      int tc = (t < T_) ? t : (T_ - 1);
      int idx = rpe[(size_t)tc * KL_ + kc];
      float w = c[i] + qr[((size_t)n * 800 + tc) * 132 + idx];
      bool dead = (kcol >= KL_) || (amask[(size_t)tc * KL_ + kc] != 0) || (kcol >= klen);
      lds[m][kcol] = dead ? NEGINF : w;   // cols 798,799 get -inf padding
    }
  }
  __syncthreads();

  // -------- phase 2: softmax stats + suppression (16 threads per row) --------
  int r = tid >> 4, ci = tid & 15;
  float mx = NEGINF;
  for (int k = ci; k < KL_; k += 16) mx = fmaxf(mx, lds[r][k]);
  red[r][ci] = mx; __syncthreads();
  if (ci == 0) { float m = red[r][0]; for (int j = 1; j < 16; ++j) m = fmaxf(m, red[r][j]); rowmax[r] = m; }
  __syncthreads();
  float M = rowmax[r];
  float s = 0.f, nzc = 0.f;
  for (int k = ci; k < KL_; k += 16) {
    float e = __expf(lds[r][k] - M);
    if (e > 0.f) { s += e; nzc += 1.f; }
  }
  red[r][ci] = s; __syncthreads();
  if (ci == 0) { float v = 0; for (int j = 0; j < 16; ++j) v += red[r][j]; rowsum[r] = v; }
  __syncthreads();
  red[r][ci] = nzc; __syncthreads();
  if (ci == 0) { float v = 0; for (int j = 0; j < 16; ++j) v += red[r][j]; rownz[r] = v; }
  __syncthreads();
  float S = rowsum[r], NZ = rownz[r];
  float invS = 1.f / S;
  float mean = 1.0f / (NZ + 1e-8f);   // key_sum == 1 after softmax
  float d = 0.f;
  for (int k = ci; k < KL_; k += 16) {
    float e = __expf(lds[r][k] - M);
    if (e > 0.f) { float p = e * invS - mean; d += p * p; }
  }
  red[r][ci] = d; __syncthreads();
  if (ci == 0) {
    float v = 0; for (int j = 0; j < 16; ++j) v += red[r][j];
    float var = v / (rownz[r] - 1.0f + 1e-8f);
    rowthr[r] = (1.0f / (rownz[r] + 1e-8f)) - 0.5f * sqrtf(var);
  }
  __syncthreads();
  float thr = rowthr[r];
  // suppress (max never suppressed: p_max >= mean >= thr) and renormalize
  float s2 = 0.f;
  for (int k = ci; k < 800; k += 16) {
    float w = lds[r][k];
    float e = __expf(w - M);
    float p = e * invS;
    if (p < thr) e = 0.f;
    lds[r][k] = e;
    s2 += e;
  }
  red[r][ci] = s2; __syncthreads();
  if (ci == 0) { float v = 0; for (int j = 0; j < 16; ++j) v += red[r][j]; rowsum2[r] = v; }
  __syncthreads();
  float inv2 = 1.f / rowsum2[r];
  for (int k = ci; k < 800; k += 16) lds[r][k] *= inv2;

  for (int j = tid; j < 16 * 132; j += 256) (&buckets[0][0])[j] = 0.f;
  for (int j = tid; j < 16 * 64;  j += 256) (&outb[0][0])[j] = 0.f;
  __syncthreads();

  // -------- phase 3: rpe-v bucket scatter (ds_add_f32) --------
  {
    int tc = t0 + r; if (tc > T_ - 1) tc = T_ - 1;
    for (int k = ci; k < KL_; k += 16) {
      float p = lds[r][k];
      if (p > 0.f) atomicAdd(&buckets[r][rpe[(size_t)tc * KL_ + k]], p);
    }
  }
  __syncthreads();

  // -------- phase 4: P @ V with WMMA (waves split K, reduce in LDS) --------
  v8f acc0 = {}, acc1 = {}, acc2 = {}, acc3 = {};
  const float* vb = vtw + ((size_t)n * 64 + l16) * 800;   // pad cols x zero probs
  for (int kb = wave; kb < 50; kb += 8) {
#pragma unroll
    for (int ks = 0; ks < 4; ++ks) {
      int kk = kb * 16 + ks * 4 + half * 2;
      v2f a = *(const v2f*)(&lds[l16][kk]);     // padded cols are zero
      v2f b0 = *(const v2f*)(vb + kk);
      v2f b1 = *(const v2f*)(vb + (size_t)16 * 800 + kk);
      v2f b2 = *(const v2f*)(vb + (size_t)32 * 800 + kk);
      v2f b3 = *(const v2f*)(vb + (size_t)48 * 800 + kk);
      acc0 = wmma_f32(a, b0, acc0);
      acc1 = wmma_f32(a, b1, acc1);
      acc2 = wmma_f32(a, b2, acc2);
      acc3 = wmma_f32(a, b3, acc3);
    }
  }
  for (int i = 0; i < 8; ++i) {
    int m = i + half * 8;
    atomicAdd(&outb[m][ 0 + l16], acc0[i]);
    atomicAdd(&outb[m][16 + l16], acc1[i]);
    atomicAdd(&outb[m][32 + l16], acc2[i]);
    atomicAdd(&outb[m][48 + l16], acc3[i]);
  }
  __syncthreads();

  // -------- phase 5: attn = 2*PV + buckets @ rpe_v_emb, write [t][b][h*64+hd] --------
  int hd = tid & 63;
  for (int rr = tid >> 6; rr < 16; rr += 4) {
    int t = t0 + rr;
    if (t < T_) {
      float v = 2.0f * outb[rr][hd];
      for (int j = 0; j < NREL; ++j) v += buckets[rr][j] * rpev[(size_t)j * 64 + hd];
      attn[(((size_t)t * B_) + b) * 512 + h * 64 + hd] = v;
    }
  }
}

// ---------------- Kernel 4: output projection + tanh on summary rows ----------------
__global__ __launch_bounds__(256) void oproj_kernel(
    const float* __restrict__ attn, const float* __restrict__ wo,
    const float* __restrict__ bo, float* __restrict__ out) {
  int wave = threadIdx.x >> 5, lane = threadIdx.x & 31;
  int half = lane >> 4, l16 = lane & 15;
  int row0 = blockIdx.x * 16;                      // rows = 798*8 = 6384 exactly
  int c0   = wave * 64 + l16;                      // 8 waves cover all 512 cols
  int r    = row0 + l16;
  const float* arow = attn + (size_t)r * 512;
  const float* brow = wo + (size_t)c0 * 512;
  v8f acc[4] = {{}, {}, {}, {}};
#pragma unroll 4
  for (int ks = 0; ks < 128; ++ks) {
    int kk = ks * 4 + half * 2;
    v2f a = *(const v2f*)(arow + kk);
    acc[0] = wmma_f32(a, *(const v2f*)(brow + kk), acc[0]);
    acc[1] = wmma_f32(a, *(const v2f*)(brow + 16 * D_ + kk), acc[1]);
    acc[2] = wmma_f32(a, *(const v2f*)(brow + 32 * D_ + kk), acc[2]);
    acc[3] = wmma_f32(a, *(const v2f*)(brow + 48 * D_ + kk), acc[3]);
  }
#pragma unroll
  for (int j = 0; j < 4; ++j) {
    int cc = c0 + j * 16;
    float bias = bo[cc];
    for (int i = 0; i < 8; ++i) {
      int rr = row0 + i + half * 8;
      int t = rr >> 3;
      float v = acc[j][i] + bias;
      if (t >= 792) v = tanhf(v);                  // next_m rows 792..797
      out[(size_t)rr * 512 + cc] = v;
    }
  }
}

extern "C" void kernel_launch(void* const* d_in, const int* in_sizes, int n_in,
                              void* d_out, int out_size, void* d_ws, size_t ws_size,
                              hipStream_t stream) {
  const float* x        = (const float*)d_in[0];
  const int*   lengths  = (const int*)d_in[1];
  const float* mems     = (const float*)d_in[2];
  const unsigned char* amask = (const unsigned char*)d_in[3];
  const int*   rpe      = (const int*)d_in[4];
  const int*   u        = (const int*)d_in[5];
  const float* wq       = (const float*)d_in[6];
  const float* bq       = (const float*)d_in[7];
  const float* wkv      = (const float*)d_in[8];
  const float* bkv      = (const float*)d_in[9];
  const float* wo       = (const float*)d_in[10];
  const float* bo       = (const float*)d_in[11];
  const float* rpek     = (const float*)d_in[12];
  const float* rpev     = (const float*)d_in[13];

  float* ws  = (float*)d_ws;
  float* qs   = ws;                                   // 64*800*64
  float* kws  = qs  + (size_t)64 * 800 * 64;          // 64*800*64
  float* vtw  = kws + (size_t)64 * 800 * 64;          // 64*64*800
  float* qr   = vtw + (size_t)64 * 800 * 64;          // 64*800*132
  float* attn = qr  + (size_t)64 * 800 * 132;         // 800*8*512

  dim3 blk(256);
  qproj_kernel <<<dim3(400),    blk, 0, stream>>>(x, wq, bq, qs);
  kvproj_kernel<<<dim3(399, 2), blk, 0, stream>>>(x, mems, wkv, bkv, kws, vtw);
  qrpe_kernel  <<<dim3(400, 9), blk, 0, stream>>>(qs, rpek, qr);
  attn_kernel  <<<dim3(50, 64), blk, 0, stream>>>(qs, kws, vtw, qr, rpe, amask,
                                                  lengths, u, rpev, attn);
  oproj_kernel <<<dim3(399),    blk, 0, stream>>>(attn, wo, bo, (float*)d_out);
}